// SparseEdgeDrop_19739669692444
// MI455X (gfx1250) — compile-verified
//
#include <hip/hip_runtime.h>
#include <hip/hip_bf16.h>
#include <stdint.h>

// val = mask ? v * 1/(1 - 0.2 + 1e-5) : 0
#define EDGE_SCALE (1.0f / (1.0f - 0.2f + 1e-5f))

typedef float f32x4 __attribute__((ext_vector_type(4)));

#define COPY_BLK 256
#define COPY_CPT 4  // 16B chunks per thread
#define COPY_STRIDE (COPY_BLK * 16)  // 4096B: per-k advance, global AND LDS

// ---------------------------------------------------------------------------
// Kernel 1: edge_index pass-through copy via the CDNA5 async global<->LDS
// datapath (GLOBAL_LOAD_ASYNC_TO_LDS_B128 / GLOBAL_STORE_ASYNC_FROM_LDS_B128,
// ASYNCcnt-tracked). Data never occupies VGPRs.
//
// Each block owns COPY_BLK*COPY_CPT contiguous 16B chunks; thread t handles
// chunks {base + k*COPY_BLK + t} -> every wave issue is a coalesced burst.
// The async ops' 24-bit IOFFSET is added to BOTH the global and LDS address
// and our layout advances both
// sides by the same 4096B per k, so interior blocks need exactly ONE address
// VGPR pair + ONE LDS-offset VGPR: 4 loads, 1 s_wait_asynccnt, 4 stores,
// zero per-chunk address math. Only the final block pays EXEC masking.
// ---------------------------------------------------------------------------
__global__ __launch_bounds__(COPY_BLK) void edge_index_copy_async(
    const uint32_t* __restrict__ src, uint32_t* __restrict__ dst,
    long long n32) {
  __shared__ alignas(16) unsigned char stage[COPY_BLK * 16 * COPY_CPT];

  const long long chunks = n32 >> 2;  // number of 16B chunks
  const long long base = (long long)blockIdx.x * (COPY_BLK * COPY_CPT);

  // Low 32 bits of the flat shared-aperture address == raw LDS offset
  // also keeps `stage` alive.
  const unsigned lds0 = (unsigned)(size_t)(void*)&stage[threadIdx.x * 16u];

  if (base + (COPY_BLK * COPY_CPT) <= chunks) {
    // ---- fast path: whole block in range, no EXEC masking ----
    const uint32_t* s = src + (base + threadIdx.x) * 4;
    uint32_t* d = dst + (base + threadIdx.x) * 4;
#pragma unroll
    for (int k = 0; k < COPY_CPT; ++k) {
      asm volatile("global_load_async_to_lds_b128 %0, %1, off offset:%2"
                   :
                   : "v"(lds0), "v"(s), "n"(k * COPY_STRIDE)
                   : "memory");
    }
    asm volatile("s_wait_asynccnt 0" ::: "memory");
#pragma unroll
    for (int k = 0; k < COPY_CPT; ++k) {
      asm volatile("global_store_async_from_lds_b128 %0, %1, off offset:%2"
                   :
                   : "v"(d), "v"(lds0), "n"(k * COPY_STRIDE)
                   : "memory");
    }
  } else {
    // ---- tail block: per-chunk guards ----
#pragma unroll
    for (int k = 0; k < COPY_CPT; ++k) {
      const long long c = base + (long long)k * COPY_BLK + threadIdx.x;
      if (c < chunks) {
        const uint32_t* s = src + c * 4;
        const unsigned l = lds0 + (unsigned)(k * COPY_STRIDE);
        asm volatile("global_load_async_to_lds_b128 %0, %1, off"
                     :
                     : "v"(l), "v"(s)
                     : "memory");
      }
    }
    asm volatile("s_wait_asynccnt 0" ::: "memory");
#pragma unroll
    for (int k = 0; k < COPY_CPT; ++k) {
      const long long c = base + (long long)k * COPY_BLK + threadIdx.x;
      if (c < chunks) {
        uint32_t* d = dst + c * 4;
        const unsigned l = lds0 + (unsigned)(k * COPY_STRIDE);
        asm volatile("global_store_async_from_lds_b128 %0, %1, off"
                     :
                     : "v"(d), "v"(l)
                     : "memory");
      }
    }
    // n32 % 4 scalar tail (never taken for this problem size), done by the
    // first threads of the (single) tail block.
    const long long tail_base = chunks * 4;
    const long long rem = n32 - tail_base;
    if ((long long)threadIdx.x < rem)
      dst[tail_base + threadIdx.x] = src[tail_base + threadIdx.x];
  }
  // S_ENDPGM waits idle anyway; explicit for clarity.
  asm volatile("s_wait_asynccnt 0" ::: "memory");
}

// ---------------------------------------------------------------------------
// Kernel 2: masked scale of edge_values, 8 edges per thread:
//   2x B128 NT loads (values), 1x B64 NT load (8 mask bytes),
//   2x B128 NT stores. Every byte is touched exactly once, so non-temporal
// hints keep this ~300MB stream from churning the 192MB L2.
// ---------------------------------------------------------------------------
__global__ __launch_bounds__(256) void edge_value_mask_scale(
    const float* __restrict__ vals, const unsigned char* __restrict__ mask,
    float* __restrict__ out, long long n8, long long n) {
  const long long i =
      (long long)blockIdx.x * (long long)blockDim.x + (long long)threadIdx.x;

  if (i < n8) {
    const f32x4 v0 = __builtin_nontemporal_load((const f32x4*)vals + 2 * i);
    const f32x4 v1 = __builtin_nontemporal_load((const f32x4*)vals + 2 * i + 1);
    const uint64_t m = __builtin_nontemporal_load((const uint64_t*)mask + i);
    f32x4 o0, o1;
    o0.x = (m & 0x00000000000000FFull) ? v0.x * EDGE_SCALE : 0.0f;
    o0.y = (m & 0x000000000000FF00ull) ? v0.y * EDGE_SCALE : 0.0f;
    o0.z = (m & 0x0000000000FF0000ull) ? v0.z * EDGE_SCALE : 0.0f;
    o0.w = (m & 0x00000000FF000000ull) ? v0.w * EDGE_SCALE : 0.0f;
    o1.x = (m & 0x000000FF00000000ull) ? v1.x * EDGE_SCALE : 0.0f;
    o1.y = (m & 0x0000FF0000000000ull) ? v1.y * EDGE_SCALE : 0.0f;
    o1.z = (m & 0x00FF000000000000ull) ? v1.z * EDGE_SCALE : 0.0f;
    o1.w = (m & 0xFF00000000000000ull) ? v1.w * EDGE_SCALE : 0.0f;
    __builtin_nontemporal_store(o0, (f32x4*)out + 2 * i);
    __builtin_nontemporal_store(o1, (f32x4*)out + 2 * i + 1);
  }

  // Scalar tail for n % 8 (not taken for E = 2^25).
  const long long tail_base = n8 * 8;
  const long long rem = n - tail_base;
  if (i < rem) {
    const long long t = tail_base + i;
    out[t] = mask[t] ? vals[t] * EDGE_SCALE : 0.0f;
  }
}

// ---------------------------------------------------------------------------
// Launch. Outputs concatenated flat in return order:
//   [ edge_index passthrough : in_sizes[0] x 4B ][ val : in_sizes[1] x f32 ]
// (int32 and float32 are both 4B, so the val offset in "elements" is
// in_sizes[0] regardless of how d_out is typed.)
// ---------------------------------------------------------------------------
extern "C" void kernel_launch(void* const* d_in, const int* in_sizes, int n_in,
                              void* d_out, int out_size, void* d_ws,
                              size_t ws_size, hipStream_t stream) {
  const uint32_t* edge_index = (const uint32_t*)d_in[0];      // int32 (2,E) flat
  const float* edge_values = (const float*)d_in[1];           // f32 (E,)
  const unsigned char* mask = (const unsigned char*)d_in[2];  // bool (E,)

  const long long nIdx = (long long)in_sizes[0];  // 2*E int32 elements
  const long long nE = (long long)in_sizes[1];    // E

  uint32_t* out_idx = (uint32_t*)d_out;
  float* out_val = (float*)d_out + nIdx;

  // Copy kernel: COPY_CPT x 16B chunks per thread.
  const long long chunks = (nIdx + 3) >> 2;  // round up so tail work exists
  const long long per_block = (long long)COPY_BLK * COPY_CPT;
  long long grid_copy = (chunks + per_block - 1) / per_block;
  if (grid_copy < 1) grid_copy = 1;
  edge_index_copy_async<<<dim3((unsigned)grid_copy), dim3(COPY_BLK), 0,
                          stream>>>(edge_index, out_idx, nIdx);

  // Value kernel: 8 edges per thread.
  const long long n8 = nE >> 3;
  long long grid_val = (n8 + 255) / 256;
  if (grid_val < 1) grid_val = 1;
  edge_value_mask_scale<<<dim3((unsigned)grid_val), dim3(256), 0, stream>>>(
      edge_values, mask, out_val, n8, nE);
}